// SparQAttention_74113955659943
// MI455X (gfx1250) — compile-verified
//
#include <hip/hip_runtime.h>

typedef __attribute__((ext_vector_type(2))) float    v2f;
typedef __attribute__((ext_vector_type(8))) float    v8f;
typedef __attribute__((ext_vector_type(4))) unsigned v4u;
typedef __attribute__((ext_vector_type(4))) int      v4i;
typedef __attribute__((ext_vector_type(8))) int      v8i;

#define S_LEN    4096
#define D_DIM    128
#define R_TOP    16
#define K_TOP    256
#define NTHREADS 512
#define NWAVES   16

__device__ __forceinline__ float waveSum(float v) {
    #pragma unroll
    for (int o = 16; o > 0; o >>= 1) v += __shfl_down(v, o, 32);
    return v;
}
__device__ __forceinline__ float waveMax(float v) {
    #pragma unroll
    for (int o = 16; o > 0; o >>= 1) v = fmaxf(v, __shfl_down(v, o, 32));
    return v;
}
// All 512 threads must call these; every thread returns the full result.
__device__ __forceinline__ float blockSum(float v, float* scr) {
    const int lane = threadIdx.x & 31, wid = threadIdx.x >> 5;
    v = waveSum(v);
    __syncthreads();                 // protect scr from previous use
    if (lane == 0) scr[wid] = v;
    __syncthreads();
    float r = scr[0];
    #pragma unroll
    for (int i = 1; i < NWAVES; ++i) r += scr[i];
    return r;
}
__device__ __forceinline__ float blockMax(float v, float* scr) {
    const int lane = threadIdx.x & 31, wid = threadIdx.x >> 5;
    v = waveMax(v);
    __syncthreads();
    if (lane == 0) scr[wid] = v;
    __syncthreads();
    float r = scr[0];
    #pragma unroll
    for (int i = 1; i < NWAVES; ++i) r = fmaxf(r, scr[i]);
    return r;
}
// Monotone float -> uint mapping (order-preserving).
__device__ __forceinline__ unsigned fkey(float f) {
    unsigned u = __float_as_uint(f);
    return (u & 0x80000000u) ? ~u : (u | 0x80000000u);
}

__global__ __launch_bounds__(NTHREADS)
void sparq_attn_kernel(const float* __restrict__ Q, const float* __restrict__ K,
                       const float* __restrict__ V, const unsigned char* __restrict__ mask,
                       float* __restrict__ out) {
    __shared__ __align__(16) float ls_q[D_DIM];
    __shared__ float ls_absq[D_DIM];
    __shared__ float ls_tv[D_DIM];
    __shared__ int   ls_ti[D_DIM];
    __shared__ int   ls_i1[R_TOP];
    __shared__ float ls_score[S_LEN];     // scaled approx scores
    __shared__ int   ls_sel[K_TOP];       // selected key indices
    __shared__ __align__(16) float ls_s[K_TOP];   // exact QK, then softmax probs
    __shared__ float ls_y[D_DIM];
    __shared__ float ls_part[4 * D_DIM];
    __shared__ float ls_scr[NWAVES];
    __shared__ float ls_scale;
    __shared__ int   ls_cnt;
    __shared__ __align__(16) unsigned char ls_maskb[S_LEN];  // mask row, TDM-loaded

    const int bh   = blockIdx.x;
    const int b    = bh >> 5;             // H = 32
    const int t    = threadIdx.x;
    const int lane = t & 31, wid = t >> 5;
    const size_t kvBase = (size_t)bh * S_LEN * D_DIM;
    const float* Kp = K + kvBase;
    const float* Vp = V + kvBase;
    const unsigned char* mp = mask + (size_t)b * S_LEN;

    // ---- TDM: DMA the 4KB mask row into LDS (Tensor Data Mover, 1x4096 u8 tile) ----
    if (wid == 0) {
        const unsigned      ldsOff = (unsigned)(unsigned long long)(const void*)&ls_maskb[0];
        const unsigned long long ga = (unsigned long long)(const void*)mp;
        v4u g0;
        g0.x = 1u;                                   // count=1 valid descriptor
        g0.y = ldsOff;                               // lds_addr
        g0.z = (unsigned)ga;                         // global_addr[31:0]
        g0.w = (unsigned)(ga >> 32) | (2u << 30);    // global_addr[56:32] | type=2
        v8i g1;
        g1[0] = 0;                                   // wg_mask=0, data_size=1B, no flags
        g1[1] = (int)(4096u << 16);                  // tensor_dim0[15:0]
        g1[2] = (int)(1u << 16);                     // tensor_dim0[31:16]=0, tensor_dim1=1
        g1[3] = (int)(4096u << 16);                  // tile_dim0 = 4096
        g1[4] = 1;                                   // tile_dim1=1, tile_dim2=0
        g1[5] = 4096;                                // tensor_dim0_stride
        g1[6] = 0;
        g1[7] = 0;
        v4i gz = {0, 0, 0, 0};
#if __clang_major__ >= 23
        v8i gz8 = {0, 0, 0, 0, 0, 0, 0, 0};
        __builtin_amdgcn_tensor_load_to_lds(g0, g1, gz, gz, gz8, 0);
#else
        __builtin_amdgcn_tensor_load_to_lds(g0, g1, gz, gz, 0);
#endif
        __builtin_amdgcn_s_wait_tensorcnt(0);
    }

    // ---- Load Q, |Q| (overlaps with TDM) ----
    if (t < D_DIM) {
        float q = Q[(size_t)bh * D_DIM + t];
        ls_q[t] = q;
        ls_absq[t] = fabsf(q);
    }
    __syncthreads();    // Q + mask both in LDS now

    float sumAbsQ = blockSum((t < D_DIM) ? ls_absq[t] : 0.f, ls_scr);

    // ---- top-r=16 of |Q| via 16 destructive tree-argmax passes ----
    float sumAbsQhat_t0 = 0.f;
    for (int it = 0; it < R_TOP; ++it) {
        if (t < D_DIM) { ls_tv[t] = ls_absq[t]; ls_ti[t] = t; }
        __syncthreads();
        for (int st = 64; st >= 1; st >>= 1) {
            if (t < st) {
                if (ls_tv[t + st] > ls_tv[t]) { ls_tv[t] = ls_tv[t + st]; ls_ti[t] = ls_ti[t + st]; }
            }
            __syncthreads();
        }
        if (t == 0) {
            int idx = ls_ti[0];
            sumAbsQhat_t0 += ls_tv[0];
            ls_i1[it] = idx;
            ls_absq[idx] = -1.f;          // remove from further passes
        }
        __syncthreads();
    }
    if (t == 0) ls_scale = sqrtf((float)D_DIM * sumAbsQhat_t0 / sumAbsQ);
    __syncthreads();
    const float invScale = 1.f / ls_scale;

    // ---- approximate scores: Q_hat . K_hat^T over all S (gathered 16 dims) ----
    int   i1r[R_TOP];
    float qh[R_TOP];
    #pragma unroll
    for (int j = 0; j < R_TOP; ++j) { i1r[j] = ls_i1[j]; qh[j] = ls_q[i1r[j]]; }

    for (int s = t; s < S_LEN; s += NTHREADS) {
        const float* krow = Kp + (size_t)s * D_DIM;
        if (s + NTHREADS < S_LEN)
            __builtin_prefetch(Kp + (size_t)(s + NTHREADS) * D_DIM, 0, 1);
        float acc = 0.f;
        #pragma unroll
        for (int j = 0; j < R_TOP; ++j) acc += qh[j] * krow[i1r[j]];
        acc *= invScale;
        if (!ls_maskb[s]) acc = -1e30f;
        ls_score[s] = acc;
    }
    __syncthreads();

    // ---- softmax stats over 4096 scaled scores ----
    float lm = -1e30f;
    for (int s = t; s < S_LEN; s += NTHREADS) lm = fmaxf(lm, ls_score[s]);
    const float smax = blockMax(lm, ls_scr);
    float le = 0.f;
    for (int s = t; s < S_LEN; s += NTHREADS) le += __expf(ls_score[s] - smax);
    const float denom = blockSum(le, ls_scr);

    // ---- top-k=256 threshold: 32-step bit-descent binary search on fkey ----
    unsigned T = 0u;
    for (int bit = 31; bit >= 0; --bit) {
        unsigned cand = T | (1u << bit);
        float c = 0.f;
        for (int s = t; s < S_LEN; s += NTHREADS)
            c += (fkey(ls_score[s]) >= cand) ? 1.f : 0.f;
        float tot = blockSum(c, ls_scr);  // same value in every thread -> T stays uniform
        if (tot >= (float)K_TOP) T = cand;
    }

    // ---- compaction: strict-greater first (provably < 256), then ties ----
    if (t == 0) ls_cnt = 0;
    __syncthreads();
    for (int s = t; s < S_LEN; s += NTHREADS)
        if (fkey(ls_score[s]) > T) { int p = atomicAdd(&ls_cnt, 1); if (p < K_TOP) ls_sel[p] = s; }
    __syncthreads();
    for (int s = t; s < S_LEN; s += NTHREADS)
        if (fkey(ls_score[s]) == T) { int p = atomicAdd(&ls_cnt, 1); if (p < K_TOP) ls_sel[p] = s; }
    __syncthreads();

    // ---- alpha = sum of top-k softmax probs ----
    float la = 0.f;
    for (int j = t; j < K_TOP; j += NTHREADS) la += __expf(ls_score[ls_sel[j]] - smax);
    const float alpha = blockSum(la, ls_scr) / denom;

    // Branch-free selectors for building the M=0-only A fragment (hoisted; no EXEC games).
    const float sel0  = (lane == 0)  ? 1.f : 0.f;
    const float sel16 = (lane == 16) ? 1.f : 0.f;
    const int   n     = lane & 15;
    const int   half  = lane >> 4;

    // ---- exact QK = Q(1x128) . K_sel^T(128x256) via WMMA f32 16x16x4 ----
    // 16 waves, wave w owns output columns [16w, 16w+16). Q occupies row M=0 of A.
    // Software-pipelined: operands for step kb+4 are fetched before the WMMA of step kb.
    {
        const int col = wid * 16 + n;
        const float* kr = Kp + (size_t)ls_sel[col] * D_DIM + 2 * half;
        v8f c = {0.f, 0.f, 0.f, 0.f, 0.f, 0.f, 0.f, 0.f};
        float4 q4 = *(const float4*)&ls_q[0];
        v2f    bf = *(const v2f*)kr;
        #pragma unroll
        for (int kb = 0; kb < D_DIM; kb += 4) {
            const int kbn = (kb + 4 < D_DIM) ? (kb + 4) : kb;   // clamped, branch-free
            const float4 q4n = *(const float4*)&ls_q[kbn];      // ds_load_b128 (next)
            const v2f    bfn = *(const v2f*)(kr + kbn);         // global_load_b64 (next)
            v2f a;  // A 16x4 f32: lane0 -> K=0,1 ; lane16 -> K=2,3 ; rows 1..15 zero
            a.x = sel0 * q4.x + sel16 * q4.z;
            a.y = sel0 * q4.y + sel16 * q4.w;
            c = __builtin_amdgcn_wmma_f32_16x16x4_f32(false, a, false, bf, (short)0, c,
                                                      false, false);
            q4 = q4n;
            bf = bfn;
        }
        // D row M=0 lives in VGPR0 of lanes 0..15
        if (lane < 16) ls_s[col] = c[0] * 0.08838834764831845f;  // 1/sqrt(128)
    }
    __syncthreads();

    // ---- softmax over the 256 exact scores ----
    const float qv = (t < K_TOP) ? ls_s[t] : -1e30f;
    const float qmax = blockMax(qv, ls_scr);
    const float pe = (t < K_TOP) ? __expf(qv - qmax) : 0.f;
    const float psum = blockSum(pe, ls_scr);
    if (t < K_TOP) ls_s[t] = pe / psum;
    __syncthreads();

    // ---- y = s(1x256) . V_sel(256x128) via WMMA f32 16x16x4 ----
    // waves 0..7 own 16-column tiles of y; uniform branch keeps EXEC all-ones per wave.
    if (wid < 8) {
        const int dbase = wid * 16;
        v8f c = {0.f, 0.f, 0.f, 0.f, 0.f, 0.f, 0.f, 0.f};
        float4 s4 = *(const float4*)&ls_s[0];
        v2f    bf;
        {
            const int r0 = ls_sel[2 * half];
            const int r1 = ls_sel[2 * half + 1];
            bf.x = Vp[(size_t)r0 * D_DIM + dbase + n];
            bf.y = Vp[(size_t)r1 * D_DIM + dbase + n];
        }
        #pragma unroll 8
        for (int kb = 0; kb < K_TOP; kb += 4) {
            const int kbn = (kb + 4 < K_TOP) ? (kb + 4) : kb;   // clamped, branch-free
            const float4 s4n = *(const float4*)&ls_s[kbn];
            const int r0 = ls_sel[kbn + 2 * half];
            const int r1 = ls_sel[kbn + 2 * half + 1];
            v2f bfn;
            bfn.x = Vp[(size_t)r0 * D_DIM + dbase + n];
            bfn.y = Vp[(size_t)r1 * D_DIM + dbase + n];
            v2f a;
            a.x = sel0 * s4.x + sel16 * s4.z;
            a.y = sel0 * s4.y + sel16 * s4.w;
            c = __builtin_amdgcn_wmma_f32_16x16x4_f32(false, a, false, bf, (short)0, c,
                                                      false, false);
            s4 = s4n;
            bf = bfn;
        }
        if (lane < 16) ls_y[dbase + n] = c[0];
    }
    __syncthreads();

    // ---- V_mean over all S (coalesced stream of V) ----
    float mcl = 0.f;
    for (int s = t; s < S_LEN; s += NTHREADS) mcl += ls_maskb[s] ? 1.f : 0.f;
    const float mcount = blockSum(mcl, ls_scr);
    {
        const int d  = t & 127;
        const int ch = t >> 7;            // 4 row-chunks of 1024
        float acc = 0.f;
        for (int s = ch * 1024; s < (ch + 1) * 1024; ++s)
            acc += ls_maskb[s] ? Vp[(size_t)s * D_DIM + d] : 0.f;
        ls_part[ch * D_DIM + d] = acc;
    }
    __syncthreads();

    // ---- output: V_mean + alpha * (y - V_mean) ----
    if (t < D_DIM) {
        float vm = (ls_part[t] + ls_part[D_DIM + t] + ls_part[2 * D_DIM + t] +
                    ls_part[3 * D_DIM + t]) / mcount;
        out[(size_t)bh * D_DIM + t] = vm + alpha * (ls_y[t] - vm);
    }
}

extern "C" void kernel_launch(void* const* d_in, const int* in_sizes, int n_in,
                              void* d_out, int out_size, void* d_ws, size_t ws_size,
                              hipStream_t stream) {
    (void)in_sizes; (void)n_in; (void)out_size; (void)d_ws; (void)ws_size;
    const float* Q = (const float*)d_in[0];
    const float* K = (const float*)d_in[1];
    const float* V = (const float*)d_in[2];
    const unsigned char* mask = (const unsigned char*)d_in[3];  // jnp.bool_ -> 1 byte/elem
    float* out = (float*)d_out;
    // One workgroup per (b,h): 8*32 = 256 workgroups, 512 threads = 16 wave32s each.
    sparq_attn_kernel<<<dim3(256), dim3(NTHREADS), 0, stream>>>(Q, K, V, mask, out);
}